// CNF_regularized_23278722744620
// MI455X (gfx1250) — compile-verified
//
#include <hip/hip_runtime.h>
#include <hip/hip_bf16.h>
#include <math.h>

// Problem constants (match the reference).
#define Bsz 2048
#define Dd  64
#define Hh  512

typedef __attribute__((ext_vector_type(2))) float v2f;
typedef __attribute__((ext_vector_type(8))) float v8f;

// One 16x16 f32 output tile via V_WMMA_F32_16X16X4_F32, K stepped by 4.
// ISA operand layout (wave32):
//   A 16x4: lanes 0-15 hold M=lane, K={0,1}; lanes 16-31 hold M=lane-16, K={2,3}
//   B 4x16: lanes 0-15 hold N=lane, K={0,1}; lanes 16-31 hold N=lane-16, K={2,3}
//   C/D 16x16: vgpr r -> M = r + 8*(lane>=16), N = lane%16
__device__ __forceinline__ v8f wmma_tile_f32(const float* __restrict__ A, int lda,
                                             const float* __restrict__ Bm, int ldb,
                                             int row0, int col0, int K, int lane)
{
    const int hf = lane >> 4;
    const int l  = lane & 15;
    const int kb = hf << 1;
    v8f acc = {};
    const float* ap = A + (size_t)(row0 + l) * lda + kb;
    const float* bp = Bm + (size_t)kb * ldb + col0 + l;
    #pragma unroll 4
    for (int k = 0; k < K; k += 4) {
        v2f a = *(const v2f*)ap;
        v2f b;
        b.x = bp[0];
        b.y = bp[ldb];
        acc = __builtin_amdgcn_wmma_f32_16x16x4_f32(false, a, false, b,
                                                    (short)0, acc, false, false);
        ap += 4;
        bp += (size_t)4 * ldb;
    }
    return acc;
}

// Precompute M[j,i] = W2[j,i] * C[j,i],  C[j,i] = sum_k W1[k,j] * W3[i,k]
// (sample-independent part of the Jacobian trace). Output 512x512.
__global__ void kP_buildM(const float* __restrict__ W1, const float* __restrict__ W2,
                          const float* __restrict__ W3, float* __restrict__ Mm)
{
    const int wave = blockIdx.x * (blockDim.x >> 5) + (threadIdx.x >> 5);
    const int lane = threadIdx.x & 31;
    const int hf = lane >> 4, l = lane & 15, kb = hf << 1;
    const int tj = wave >> 5, ti = wave & 31;      // 32 x 32 tiles
    const int j0 = tj * 16, i0 = ti * 16;
    v8f acc = {};
    #pragma unroll 4
    for (int k = 0; k < Dd; k += 4) {
        v2f a;                                      // A(m,k) = W1[k, j0+m] (transposed read)
        a.x = W1[(size_t)(k + kb    ) * Hh + j0 + l];
        a.y = W1[(size_t)(k + kb + 1) * Hh + j0 + l];
        v2f b = *(const v2f*)&W3[(size_t)(i0 + l) * Dd + k + kb];  // B(k,n) = W3[i0+n, k]
        acc = __builtin_amdgcn_wmma_f32_16x16x4_f32(false, a, false, b,
                                                    (short)0, acc, false, false);
    }
    #pragma unroll
    for (int r = 0; r < 8; ++r) {
        const int j = j0 + r + 8 * hf;
        const int i = i0 + l;
        Mm[(size_t)j * Hh + i] = acc[r] * W2[(size_t)j * Hh + i];
    }
}

// Layer 1: H1 = tanh(Z @ W1[0:64,:] + (b1 + t*W1[64,:])), S1 = 1 - H1^2
__global__ void k1_layer1(const float* __restrict__ Z, const float* __restrict__ W1,
                          const float* __restrict__ b1, const float* __restrict__ t,
                          float* __restrict__ H1, float* __restrict__ S1)
{
    const int wave = blockIdx.x * (blockDim.x >> 5) + (threadIdx.x >> 5);
    const int lane = threadIdx.x & 31;
    const int tm = wave >> 5, tn = wave & 31;      // 128 x 32 tiles
    const int row0 = tm * 16, col0 = tn * 16;
    v8f acc = wmma_tile_f32(Z, Dd, W1, Hh, row0, col0, Dd, lane);
    const int hf = lane >> 4, l = lane & 15;
    const int col = col0 + l;
    const float bias = b1[col] + t[0] * W1[(size_t)Dd * Hh + col];
    #pragma unroll
    for (int r = 0; r < 8; ++r) {
        const int row = row0 + r + 8 * hf;
        const float h = tanhf(acc[r] + bias);
        H1[(size_t)row * Hh + col] = h;
        S1[(size_t)row * Hh + col] = 1.0f - h * h;
    }
}

// Layer 2: H2 = tanh(H1 @ W2 + b2), S2 = 1 - H2^2.
// Largest GEMM ([2048x512]@[512x512]) -> 2x2 register tiling: each wave owns a
// 32x32 output block (4 WMMA tiles), reusing each A/B fragment twice
// (6 load instructions per 4 WMMAs instead of 12).
__global__ void k2_layer2(const float* __restrict__ H1, const float* __restrict__ W2,
                          const float* __restrict__ b2,
                          float* __restrict__ H2, float* __restrict__ S2)
{
    const int wave = blockIdx.x * (blockDim.x >> 5) + (threadIdx.x >> 5);
    const int lane = threadIdx.x & 31;
    const int hf = lane >> 4, l = lane & 15, kb = hf << 1;
    const int tm = wave >> 4, tn = wave & 15;      // 64 x 16 supertiles of 32x32
    const int row0 = tm * 32, col0 = tn * 32;

    v8f a00 = {}, a01 = {}, a10 = {}, a11 = {};
    const float* ap0 = H1 + (size_t)(row0 + l) * Hh + kb;
    const float* ap1 = ap0 + (size_t)16 * Hh;
    const float* bp0 = W2 + (size_t)kb * Hh + col0 + l;
    const float* bp1 = bp0 + 16;
    #pragma unroll 2
    for (int k = 0; k < Hh; k += 4) {
        v2f fa0 = *(const v2f*)ap0;
        v2f fa1 = *(const v2f*)ap1;
        v2f fb0; fb0.x = bp0[0]; fb0.y = bp0[Hh];
        v2f fb1; fb1.x = bp1[0]; fb1.y = bp1[Hh];
        a00 = __builtin_amdgcn_wmma_f32_16x16x4_f32(false, fa0, false, fb0, (short)0, a00, false, false);
        a01 = __builtin_amdgcn_wmma_f32_16x16x4_f32(false, fa0, false, fb1, (short)0, a01, false, false);
        a10 = __builtin_amdgcn_wmma_f32_16x16x4_f32(false, fa1, false, fb0, (short)0, a10, false, false);
        a11 = __builtin_amdgcn_wmma_f32_16x16x4_f32(false, fa1, false, fb1, (short)0, a11, false, false);
        ap0 += 4; ap1 += 4;
        bp0 += (size_t)4 * Hh; bp1 += (size_t)4 * Hh;
    }

    const int c0 = col0 + l, c1 = col0 + 16 + l;
    const float bias0 = b2[c0], bias1 = b2[c1];
    #pragma unroll
    for (int r = 0; r < 8; ++r) {
        const int r0 = row0 + r + 8 * hf;
        const int r1 = r0 + 16;
        float h;
        h = tanhf(a00[r] + bias0); H2[(size_t)r0 * Hh + c0] = h; S2[(size_t)r0 * Hh + c0] = 1.0f - h * h;
        h = tanhf(a01[r] + bias1); H2[(size_t)r0 * Hh + c1] = h; S2[(size_t)r0 * Hh + c1] = 1.0f - h * h;
        h = tanhf(a10[r] + bias0); H2[(size_t)r1 * Hh + c0] = h; S2[(size_t)r1 * Hh + c0] = 1.0f - h * h;
        h = tanhf(a11[r] + bias1); H2[(size_t)r1 * Hh + c1] = h; S2[(size_t)r1 * Hh + c1] = 1.0f - h * h;
    }
}

// Output layer: dz_dt = H2 @ W3 + b3  -> d_out[0 : B*D], row-major [B, D]
__global__ void k3_out(const float* __restrict__ H2, const float* __restrict__ W3,
                       const float* __restrict__ b3, float* __restrict__ out)
{
    const int wave = blockIdx.x * (blockDim.x >> 5) + (threadIdx.x >> 5);
    const int lane = threadIdx.x & 31;
    const int tm = wave >> 2, tn = wave & 3;       // 128 x 4 tiles
    const int row0 = tm * 16, col0 = tn * 16;
    v8f acc = wmma_tile_f32(H2, Hh, W3, Dd, row0, col0, Hh, lane);
    const int hf = lane >> 4, l = lane & 15;
    const int col = col0 + l;
    const float bias = b3[col];
    #pragma unroll
    for (int r = 0; r < 8; ++r) {
        const int row = row0 + r + 8 * hf;
        out[(size_t)row * Dd + col] = acc[r] + bias;
    }
}

// Trace stage A: one wave per 16x16 tile of T = S1 @ M (4096 waves).
// Fuses elementwise * S2 and in-wave lane reduction; writes per-(coltile,row)
// partial sums to workspace (no float atomics -> bitwise deterministic).
__global__ void k4a_partial(const float* __restrict__ S1, const float* __restrict__ S2,
                            const float* __restrict__ Mm, float* __restrict__ part)
{
    const int wave = blockIdx.x * (blockDim.x >> 5) + (threadIdx.x >> 5);
    const int lane = threadIdx.x & 31;
    const int hf = lane >> 4, l = lane & 15;
    const int tm = wave >> 5, nt = wave & 31;      // 128 row tiles x 32 col tiles
    const int row0 = tm * 16, col0 = nt * 16;

    v8f acc = wmma_tile_f32(S1, Hh, Mm, Hh, row0, col0, Hh, lane);

    float red[8];
    #pragma unroll
    for (int r = 0; r < 8; ++r) {
        const int row = row0 + r + 8 * hf;
        red[r] = acc[r] * S2[(size_t)row * Hh + col0 + l];
    }
    #pragma unroll
    for (int r = 0; r < 8; ++r) {
        float v = red[r];
        v += __shfl_xor(v, 1, 32);
        v += __shfl_xor(v, 2, 32);
        v += __shfl_xor(v, 4, 32);
        v += __shfl_xor(v, 8, 32);
        red[r] = v;
    }
    if (l == 0) {
        #pragma unroll
        for (int r = 0; r < 8; ++r)
            part[(size_t)nt * Bsz + row0 + r + 8 * hf] = red[r];
    }
}

// Trace stage B: deterministic reduction over the 32 column-tile partials.
__global__ void k4b_reduce(const float* __restrict__ part, float* __restrict__ tr)
{
    const int b = blockIdx.x * blockDim.x + threadIdx.x;   // 2048 rows
    float s = 0.0f;
    #pragma unroll
    for (int nt = 0; nt < Hh / 16; ++nt)
        s += part[(size_t)nt * Bsz + b];
    tr[b] = -s;
}

extern "C" void kernel_launch(void* const* d_in, const int* in_sizes, int n_in,
                              void* d_out, int out_size, void* d_ws, size_t ws_size,
                              hipStream_t stream)
{
    (void)in_sizes; (void)n_in; (void)out_size; (void)ws_size;
    // setup_inputs order: z, logp_z, t, W1, b1, W2, b2, W3, b3 (all float32)
    const float* z  = (const float*)d_in[0];
    const float* t  = (const float*)d_in[2];
    const float* W1 = (const float*)d_in[3];
    const float* b1 = (const float*)d_in[4];
    const float* W2 = (const float*)d_in[5];
    const float* b2 = (const float*)d_in[6];
    const float* W3 = (const float*)d_in[7];
    const float* b3 = (const float*)d_in[8];
    float* out = (float*)d_out;

    // Workspace: H1,S1,H2,S2 (each B*H f32), M (H*H f32), partials (32*B f32)
    float* H1 = (float*)d_ws;
    float* S1 = H1 + (size_t)Bsz * Hh;
    float* H2 = S1 + (size_t)Bsz * Hh;
    float* S2 = H2 + (size_t)Bsz * Hh;
    float* Mm = S2 + (size_t)Bsz * Hh;
    float* Pt = Mm + (size_t)Hh * Hh;

    // 256 threads = 8 wave32 per block.
    kP_buildM  <<<128, 256, 0, stream>>>(W1, W2, W3, Mm);                 // 1024 waves
    k1_layer1  <<<512, 256, 0, stream>>>(z, W1, b1, t, H1, S1);           // 4096 waves
    k2_layer2  <<<128, 256, 0, stream>>>(H1, W2, b2, H2, S2);             // 1024 waves (2x2 tiles)
    k3_out     <<< 64, 256, 0, stream>>>(H2, W3, b3, out);                //  512 waves
    k4a_partial<<<512, 256, 0, stream>>>(S1, S2, Mm, Pt);                 // 4096 waves
    k4b_reduce <<<  8, 256, 0, stream>>>(Pt, out + (size_t)Bsz * Dd);     // 2048 threads
}